// DUB_3324304687509
// MI455X (gfx1250) — compile-verified
//
#include <hip/hip_runtime.h>
#include <math.h>

typedef __attribute__((ext_vector_type(2))) float v2f;
typedef __attribute__((ext_vector_type(8))) float v8f;

#define EPS_BN   1e-5f
#define A_CHN    128
#define B_CHN    64
#define C_CHN    64
#define AB_CHN   192
#define BATCH_N  8
#define HH       64
#define WW       64
#define HW       4096      // 64*64
#define K3       1152      // 128*9
#define K1       192
#define KC       288       // K3/4 : im2col K-chunk (LDS = 2*288*16*4 = 36.8KB)

__device__ __forceinline__ float sigm(float v) { return 1.0f / (1.0f + __expf(-v)); }

// ---------------------------------------------------------------------------
// Kernel 0: zero-padded copy of w_off3 (27,1152) -> wpad (32,1152)
// ---------------------------------------------------------------------------
__global__ void pad_w_kernel(const float* __restrict__ w_off3, float* __restrict__ wpad)
{
    int idx = blockIdx.x * 256 + threadIdx.x;           // 32*1152 = 36864
    int m = idx / K3;
    wpad[idx] = (m < 27) ? w_off3[idx] : 0.0f;
}

// ---------------------------------------------------------------------------
// Kernel 1: h = bn_relu(x, bn3); h1[:, 0:128] = bn_relu(x, bn1[0:128])
// ---------------------------------------------------------------------------
__global__ void bn_relu_x_kernel(const float* __restrict__ x,
                                 const float* __restrict__ g3, const float* __restrict__ be3,
                                 const float* __restrict__ mu3, const float* __restrict__ va3,
                                 const float* __restrict__ g1, const float* __restrict__ be1,
                                 const float* __restrict__ mu1, const float* __restrict__ va1,
                                 float* __restrict__ h, float* __restrict__ h1)
{
    int idx = blockIdx.x * 256 + threadIdx.x;           // 8*128*4096 total
    int p = idx & (HW - 1);
    int c = (idx >> 12) & 127;
    int b = idx >> 19;
    float xv = x[idx];
    float s3 = g3[c] * rsqrtf(va3[c] + EPS_BN);
    float o3 = be3[c] - mu3[c] * s3;
    h[idx] = fmaxf(xv * s3 + o3, 0.0f);
    float s1 = g1[c] * rsqrtf(va1[c] + EPS_BN);
    float o1 = be1[c] - mu1[c] * s1;
    h1[((size_t)b * AB_CHN + c) * HW + p] = fmaxf(xv * s1 + o1, 0.0f);
}

// ---------------------------------------------------------------------------
// Kernel 2: omm = conv3x3(h, wpad) + b_off3      (27 out channels, M padded 32)
// Implicit-GEMM: M=32, K=1152, N=32 pixels / workgroup. Waves 0-3 do WMMA.
// ---------------------------------------------------------------------------
__global__ void off_conv3_kernel(const float* __restrict__ h,
                                 const float* __restrict__ wpad,     // (32,1152) padded
                                 const float* __restrict__ b_off3,
                                 float* __restrict__ omm)             // (8,27,64,64)
{
    __shared__ float cols[2 * KC * 16];                 // [ntile][kloc][16], 36.8 KB
    int blk = blockIdx.x;
    int b   = blk >> 7;
    int rem = blk & 127;
    int ho  = rem >> 1;
    int wo0 = (rem & 1) * 32;
    int tid = threadIdx.x;

    const float* hb = h + (size_t)b * A_CHN * HW;

    // GEMM lane constants: waves 0-3 -> (mtile = w&1, ntile = w>>1)
    int w     = tid >> 5;
    int l     = tid & 31;
    int n     = l & 15;
    int koff  = (l >> 4) * 2;
    int mtile = w & 1;
    int ntile = w >> 1;
    int m     = mtile * 16 + n;
    const float* Arow  = wpad + (size_t)m * K3;
    const float* Bbase = cols + ntile * (KC * 16);
    v8f acc = {};

    for (int chunk = 0; chunk < 4; ++chunk) {
        int base = chunk * KC;
        __syncthreads();
        // build im2col chunk: KC*32 elems / 256 threads = 36 iters
        for (int it = 0; it < 36; ++it) {
            int idx  = it * 256 + tid;
            int pix  = idx & 31;
            int kloc = idx >> 5;
            int k  = base + kloc;
            int c  = k / 9;
            int kk = k - c * 9;
            int ky = kk / 3, kx = kk - ky * 3;
            int y = ho - 1 + ky;
            int x = wo0 + pix - 1 + kx;
            float v = 0.0f;
            if (y >= 0 && y < HH && x >= 0 && x < WW)
                v = hb[c * HW + y * WW + x];
            cols[(pix >> 4) * (KC * 16) + kloc * 16 + (pix & 15)] = v;
        }
        __syncthreads();
        if (tid < 128) {
            for (int k0 = 0; k0 < KC; k0 += 4) {
                v2f a = *(const v2f*)(Arow + base + k0 + koff);
                v2f bb;
                bb.x = Bbase[(k0 + koff) * 16 + n];
                bb.y = Bbase[(k0 + koff + 1) * 16 + n];
                acc = __builtin_amdgcn_wmma_f32_16x16x4_f32(
                        false, a, false, bb, (short)0, acc, false, false);
            }
        }
    }
    if (tid < 128) {
        int mbase = mtile * 16 + ((l >= 16) ? 8 : 0);
        int sp = ho * WW + wo0 + ntile * 16 + n;
        for (int r = 0; r < 8; ++r) {
            int mm = mbase + r;
            if (mm < 27)
                omm[((size_t)b * 27 + mm) * HW + sp] = acc[r] + b_off3[mm];
        }
    }
}

// ---------------------------------------------------------------------------
// Kernel 3: deformable conv 3x3 (modulated) + fused bn_relu -> h1[:,128:192]
// M=64, K=1152, N=32 / workgroup; all 8 waves do WMMA (mtile = w&3, ntile = w>>2).
// ---------------------------------------------------------------------------
__global__ void dcn3_kernel(const float* __restrict__ h,
                            const float* __restrict__ omm,      // (8,27,64,64)
                            const float* __restrict__ w3,       // (64,128,3,3)
                            const float* __restrict__ b3,
                            const float* __restrict__ g1, const float* __restrict__ be1,
                            const float* __restrict__ mu1, const float* __restrict__ va1,
                            float* __restrict__ h1)
{
    __shared__ float cols[2 * KC * 16];                 // 36.8 KB
    __shared__ int   ci[288][4];                        // 9*32 corner flat idx (4.6 KB)
    __shared__ float cw[288][4];                        // weights, mask*valid folded

    int blk = blockIdx.x;
    int b   = blk >> 7;
    int rem = blk & 127;
    int ho  = rem >> 1;
    int wo0 = (rem & 1) * 32;
    int tid = threadIdx.x;

    // Phase A: per-(kk,pixel) bilinear coordinates (shared by all 128 channels)
    for (int s = tid; s < 288; s += 256) {
        int kk  = s >> 5, pix = s & 31;
        int ky  = kk / 3, kx = kk - ky * 3;
        int sp  = ho * WW + wo0 + pix;
        const float* ob = omm + (size_t)b * 27 * HW;
        float oy = ob[kk * HW + sp];
        float ox = ob[(9 + kk) * HW + sp];
        float mk = sigm(ob[(18 + kk) * HW + sp]);
        float py = (float)(ho - 1 + ky) + oy;
        float px = (float)(wo0 + pix - 1 + kx) + ox;
        float y0f = floorf(py), x0f = floorf(px);
        float wy1 = py - y0f,  wx1 = px - x0f;
        float wy0 = 1.0f - wy1, wx0 = 1.0f - wx1;
        int y0 = (int)y0f, x0 = (int)x0f;
        int y1 = y0 + 1,   x1 = x0 + 1;
        bool vy0 = (y0 >= 0) && (y0 < HH), vy1 = (y1 >= 0) && (y1 < HH);
        bool vx0 = (x0 >= 0) && (x0 < WW), vx1 = (x1 >= 0) && (x1 < WW);
        int yc0 = min(max(y0, 0), HH - 1), yc1 = min(max(y1, 0), HH - 1);
        int xc0 = min(max(x0, 0), WW - 1), xc1 = min(max(x1, 0), WW - 1);
        ci[s][0] = yc0 * WW + xc0;
        ci[s][1] = yc0 * WW + xc1;
        ci[s][2] = yc1 * WW + xc0;
        ci[s][3] = yc1 * WW + xc1;
        cw[s][0] = (vy0 && vx0) ? wy0 * wx0 * mk : 0.0f;
        cw[s][1] = (vy0 && vx1) ? wy0 * wx1 * mk : 0.0f;
        cw[s][2] = (vy1 && vx0) ? wy1 * wx0 * mk : 0.0f;
        cw[s][3] = (vy1 && vx1) ? wy1 * wx1 * mk : 0.0f;
    }

    const float* hb = h + (size_t)b * A_CHN * HW;

    int w     = tid >> 5;
    int l     = tid & 31;
    int n     = l & 15;
    int koff  = (l >> 4) * 2;
    int mtile = w & 3;
    int ntile = w >> 2;
    int m     = mtile * 16 + n;
    const float* Arow  = w3 + (size_t)m * K3;
    const float* Bbase = cols + ntile * (KC * 16);
    v8f acc = {};

    for (int chunk = 0; chunk < 4; ++chunk) {
        int base = chunk * KC;
        __syncthreads();
        // build modulated im2col chunk: KC*32/256 = 36 iters
        for (int it = 0; it < 36; ++it) {
            int idx  = it * 256 + tid;
            int pix  = idx & 31;
            int kloc = idx >> 5;
            int k  = base + kloc;
            int c  = k / 9;
            int kk = k - c * 9;
            int s  = kk * 32 + pix;
            const float* hc = hb + c * HW;
            float v = cw[s][0] * hc[ci[s][0]]
                    + cw[s][1] * hc[ci[s][1]]
                    + cw[s][2] * hc[ci[s][2]]
                    + cw[s][3] * hc[ci[s][3]];
            cols[(pix >> 4) * (KC * 16) + kloc * 16 + (pix & 15)] = v;
        }
        __syncthreads();
        for (int k0 = 0; k0 < KC; k0 += 4) {
            v2f a = *(const v2f*)(Arow + base + k0 + koff);
            v2f bb;
            bb.x = Bbase[(k0 + koff) * 16 + n];
            bb.y = Bbase[(k0 + koff + 1) * 16 + n];
            acc = __builtin_amdgcn_wmma_f32_16x16x4_f32(
                    false, a, false, bb, (short)0, acc, false, false);
        }
    }
    // Epilogue: + b3, bn_relu with bn1[128+o], write y-part of h1
    {
        int mbase = mtile * 16 + ((l >= 16) ? 8 : 0);
        int sp = ho * WW + wo0 + ntile * 16 + n;
        for (int r = 0; r < 8; ++r) {
            int o  = mbase + r;
            int ch = 128 + o;
            float s1 = g1[ch] * rsqrtf(va1[ch] + EPS_BN);
            float sh = be1[ch] - mu1[ch] * s1;
            float val = acc[r] + b3[o];
            h1[((size_t)b * AB_CHN + ch) * HW + sp] = fmaxf(val * s1 + sh, 0.0f);
        }
    }
}

// ---------------------------------------------------------------------------
// Kernel 4: 1x1 modulated deformable conv on h1 + nearest 2x upsample -> out
// M=64, K=192, N=32 / workgroup; all 8 waves do WMMA.
// ---------------------------------------------------------------------------
__global__ void dcn1_up_kernel(const float* __restrict__ h1,
                               const float* __restrict__ w_off1,  // (3,192,1,1)
                               const float* __restrict__ b_off1,
                               const float* __restrict__ w1,      // (64,192,1,1)
                               const float* __restrict__ b1,
                               float* __restrict__ out)            // (8,64,128,128)
{
    __shared__ float col[2 * K1 * 16];                  // 24.6 KB
    __shared__ float omm1[3 * 32];
    __shared__ int   ci[32][4];
    __shared__ float cw[32][4];

    int blk = blockIdx.x;
    int b   = blk >> 7;
    int rem = blk & 127;
    int ho  = rem >> 1;
    int wo0 = (rem & 1) * 32;
    int tid = threadIdx.x;

    const float* hb = h1 + (size_t)b * AB_CHN * HW;

    // Phase A: 1x1 offset conv (3 channels x 32 pixels, dot of 192)
    if (tid < 96) {
        int j = tid >> 5, pix = tid & 31;
        int sp = ho * WW + wo0 + pix;
        float s = b_off1[j];
        const float* wr = w_off1 + j * K1;
        for (int c = 0; c < K1; ++c) s += wr[c] * hb[c * HW + sp];
        omm1[j * 32 + pix] = s;
    }
    __syncthreads();

    // Phase B: bilinear coordinates (one per pixel; base = (ho, wo), no pad)
    if (tid < 32) {
        int pix = tid;
        float oy = omm1[pix], ox = omm1[32 + pix];
        float mk = sigm(omm1[64 + pix]);
        float py = (float)ho + oy;
        float px = (float)(wo0 + pix) + ox;
        float y0f = floorf(py), x0f = floorf(px);
        float wy1 = py - y0f,  wx1 = px - x0f;
        float wy0 = 1.0f - wy1, wx0 = 1.0f - wx1;
        int y0 = (int)y0f, x0 = (int)x0f;
        int y1 = y0 + 1,   x1 = x0 + 1;
        bool vy0 = (y0 >= 0) && (y0 < HH), vy1 = (y1 >= 0) && (y1 < HH);
        bool vx0 = (x0 >= 0) && (x0 < WW), vx1 = (x1 >= 0) && (x1 < WW);
        int yc0 = min(max(y0, 0), HH - 1), yc1 = min(max(y1, 0), HH - 1);
        int xc0 = min(max(x0, 0), WW - 1), xc1 = min(max(x1, 0), WW - 1);
        ci[pix][0] = yc0 * WW + xc0;
        ci[pix][1] = yc0 * WW + xc1;
        ci[pix][2] = yc1 * WW + xc0;
        ci[pix][3] = yc1 * WW + xc1;
        cw[pix][0] = (vy0 && vx0) ? wy0 * wx0 * mk : 0.0f;
        cw[pix][1] = (vy0 && vx1) ? wy0 * wx1 * mk : 0.0f;
        cw[pix][2] = (vy1 && vx0) ? wy1 * wx0 * mk : 0.0f;
        cw[pix][3] = (vy1 && vx1) ? wy1 * wx1 * mk : 0.0f;
    }
    __syncthreads();

    // Phase C: build modulated col (K=192, 32 pixels): 192*32/256 = 24 iters
    for (int it = 0; it < 24; ++it) {
        int idx = it * 256 + tid;
        int pix = idx & 31;
        int c   = idx >> 5;
        const float* hc = hb + c * HW;
        float v = cw[pix][0] * hc[ci[pix][0]]
                + cw[pix][1] * hc[ci[pix][1]]
                + cw[pix][2] * hc[ci[pix][2]]
                + cw[pix][3] * hc[ci[pix][3]];
        col[(pix >> 4) * (K1 * 16) + c * 16 + (pix & 15)] = v;
    }
    __syncthreads();

    // Phase D: GEMM M=64, K=192, N=32 + bias + 2x nearest upsample store
    {
        int w     = tid >> 5;
        int l     = tid & 31;
        int n     = l & 15;
        int koff  = (l >> 4) * 2;
        int mtile = w & 3;
        int ntile = w >> 2;
        int m     = mtile * 16 + n;
        const float* Arow  = w1 + (size_t)m * K1;
        const float* Bbase = col + ntile * (K1 * 16);
        v8f acc = {};
        for (int k0 = 0; k0 < K1; k0 += 4) {
            v2f a = *(const v2f*)(Arow + k0 + koff);
            v2f bb;
            bb.x = Bbase[(k0 + koff) * 16 + n];
            bb.y = Bbase[(k0 + koff + 1) * 16 + n];
            acc = __builtin_amdgcn_wmma_f32_16x16x4_f32(
                    false, a, false, bb, (short)0, acc, false, false);
        }
        int mbase = mtile * 16 + ((l >= 16) ? 8 : 0);
        int oy2 = 2 * ho;
        int ox2 = 2 * (wo0 + ntile * 16 + n);
        for (int r = 0; r < 8; ++r) {
            int o = mbase + r;
            float z = acc[r] + b1[o];
            float* ob = out + ((size_t)b * C_CHN + o) * (128 * 128);
            ob[oy2 * 128 + ox2]           = z;
            ob[oy2 * 128 + ox2 + 1]       = z;
            ob[(oy2 + 1) * 128 + ox2]     = z;
            ob[(oy2 + 1) * 128 + ox2 + 1] = z;
        }
    }
}

// ---------------------------------------------------------------------------
extern "C" void kernel_launch(void* const* d_in, const int* in_sizes, int n_in,
                              void* d_out, int out_size, void* d_ws, size_t ws_size,
                              hipStream_t stream)
{
    const float* x      = (const float*)d_in[0];
    const float* bn3_g  = (const float*)d_in[1];
    const float* bn3_b  = (const float*)d_in[2];
    const float* bn3_m  = (const float*)d_in[3];
    const float* bn3_v  = (const float*)d_in[4];
    const float* w_off3 = (const float*)d_in[5];
    const float* b_off3 = (const float*)d_in[6];
    const float* w3     = (const float*)d_in[7];
    const float* b3     = (const float*)d_in[8];
    const float* bn1_g  = (const float*)d_in[9];
    const float* bn1_b  = (const float*)d_in[10];
    const float* bn1_m  = (const float*)d_in[11];
    const float* bn1_v  = (const float*)d_in[12];
    const float* w_off1 = (const float*)d_in[13];
    const float* b_off1 = (const float*)d_in[14];
    const float* w1     = (const float*)d_in[15];
    const float* b1     = (const float*)d_in[16];

    float* h    = (float*)d_ws;                              // 8*128*4096
    float* h1   = h   + (size_t)BATCH_N * A_CHN  * HW;       // 8*192*4096
    float* omm  = h1  + (size_t)BATCH_N * AB_CHN * HW;       // 8*27*4096
    float* wpad = omm + (size_t)BATCH_N * 27 * HW;           // 32*1152

    pad_w_kernel<<<(32 * K3) / 256, 256, 0, stream>>>(w_off3, wpad);

    bn_relu_x_kernel<<<(BATCH_N * A_CHN * HW) / 256, 256, 0, stream>>>(
        x, bn3_g, bn3_b, bn3_m, bn3_v, bn1_g, bn1_b, bn1_m, bn1_v, h, h1);

    off_conv3_kernel<<<BATCH_N * HW / 32, 256, 0, stream>>>(h, wpad, b_off3, omm);

    dcn3_kernel<<<BATCH_N * HW / 32, 256, 0, stream>>>(
        h, omm, w3, b3, bn1_g, bn1_b, bn1_m, bn1_v, h1);

    dcn1_up_kernel<<<BATCH_N * HW / 32, 256, 0, stream>>>(
        h1, w_off1, b_off1, w1, b1, (float*)d_out);
}